// BertWithAdaThresholdLocContextPooling_47674136986015
// MI455X (gfx1250) — compile-verified
//
#include <hip/hip_runtime.h>
#include <hip/hip_bf16.h>
#include <math.h>

typedef __attribute__((ext_vector_type(2))) float v2f;
typedef __attribute__((ext_vector_type(8))) float v8f;

#define Bsz   16
#define Lsz   512
#define HIDs  768
#define HEADSn 12
#define Mn    4
#define EMBs  768
#define BLKs  8
#define NERs  6
#define NCLSs 97
#define KCAT  1542          // 2*HID + NER
#define KPAD  1544          // padded to multiple of 4
#define KBL   6144          // EMB*BLK
#define KWAVES 8            // waves per output tile (K-split)
#define PF    4             // prefetch pipeline depth (load pairs in flight)

// ---------------------------------------------------------------------------
// Kernel 1: gather mention embeddings + logsumexp over M=4 -> Xh[:,0:768], Xt[:,0:768]
// ---------------------------------------------------------------------------
__global__ void k_entity_emb(const float* __restrict__ seq,
                             const int*   __restrict__ entity_pos,
                             float* __restrict__ Xh, float* __restrict__ Xt) {
  int idx = blockIdx.x * blockDim.x + threadIdx.x;   // B*2*768
  if (idx >= Bsz * 2 * HIDs) return;
  int d = idx % HIDs;
  int e = (idx / HIDs) % 2;
  int b = idx / (HIDs * 2);
  const int* ep = entity_pos + (b * 2 + e) * Mn;
  float vals[Mn];
  float mx = -INFINITY;
#pragma unroll
  for (int m = 0; m < Mn; ++m) {
    int s = ep[m] + 1;
    vals[m] = seq[((long)b * Lsz + s) * HIDs + d];
    mx = fmaxf(mx, vals[m]);
  }
  float ssum = 0.f;
#pragma unroll
  for (int m = 0; m < Mn; ++m) ssum += expf(vals[m] - mx);
  float r = mx + logf(ssum);
  float* X = (e == 0) ? Xh : Xt;
  X[b * KPAD + d] = r;
}

// ---------------------------------------------------------------------------
// Kernel 2: pooled attention weights, normalized per batch. One block per b.
// ---------------------------------------------------------------------------
__global__ void k_ht_att(const float* __restrict__ attn,
                         const int*   __restrict__ entity_pos,
                         float* __restrict__ htn) {
  int b = blockIdx.x;
  int tid = threadIdx.x;                 // 256 threads, 2 l's each
  __shared__ float sbuf[256];
  const int* ep = entity_pos + b * 2 * Mn;
  int s0[Mn], s1[Mn];
#pragma unroll
  for (int m = 0; m < Mn; ++m) { s0[m] = ep[m] + 1; s1[m] = ep[Mn + m] + 1; }
  float vals[2];
  float local = 0.f;
#pragma unroll
  for (int j = 0; j < 2; ++j) {
    int l = tid + j * 256;
    float acc = 0.f;
    for (int h = 0; h < HEADSn; ++h) {
      const float* base = attn + ((long)(b * HEADSn + h) * Lsz) * Lsz;
      float ha = 0.f, ta = 0.f;
#pragma unroll
      for (int m = 0; m < Mn; ++m) {
        ha += base[s0[m] * Lsz + l];
        ta += base[s1[m] * Lsz + l];
      }
      acc += (ha * 0.25f) * (ta * 0.25f);
    }
    acc *= (1.0f / 12.0f);
    vals[j] = acc;
    local += acc;
  }
  sbuf[tid] = local;
  __syncthreads();
  for (int s = 128; s > 0; s >>= 1) {
    if (tid < s) sbuf[tid] += sbuf[tid + s];
    __syncthreads();
  }
  float denom = sbuf[0] + 1e-5f;
#pragma unroll
  for (int j = 0; j < 2; ++j)
    htn[b * Lsz + tid + j * 256] = vals[j] / denom;
}

// ---------------------------------------------------------------------------
// Kernel 3: rs[b,d] = sum_l seq[b,l,d]*htn[b,l]  -> Xh/Xt[:,768:1536]; NER + pad
//   Each thread owns 2 consecutive d's (float2 column), unrolled over l.
// ---------------------------------------------------------------------------
__global__ void k_rs(const float* __restrict__ seq, const float* __restrict__ htn,
                     const float* __restrict__ hs_ner, const float* __restrict__ ts_ner,
                     float* __restrict__ Xh, float* __restrict__ Xt) {
  int idx = blockIdx.x * blockDim.x + threadIdx.x;   // 16 * 384
  if (idx >= Bsz * (HIDs / 2)) return;
  int d2 = idx % (HIDs / 2), b = idx / (HIDs / 2);
  int d = d2 * 2;
  const float* hb = htn + b * Lsz;
  const float* sb = seq + (long)b * Lsz * HIDs + d;
  float acc0 = 0.f, acc1 = 0.f;
#pragma unroll 8
  for (int l = 0; l < Lsz; ++l) {
    v2f s = *(const v2f*)(sb + (long)l * HIDs);
    float hv = hb[l];
    acc0 += s.x * hv;
    acc1 += s.y * hv;
  }
  Xh[b * KPAD + HIDs + d]     = acc0;
  Xh[b * KPAD + HIDs + d + 1] = acc1;
  Xt[b * KPAD + HIDs + d]     = acc0;
  Xt[b * KPAD + HIDs + d + 1] = acc1;
  if (d < 8) {  // ner tags (6) + zero pad (2) to complete K=1544
#pragma unroll
    for (int q = 0; q < 2; ++q) {
      int dd = d + q;
      float hv = (dd < NERs) ? hs_ner[b * NERs + dd] : 0.f;
      float tv = (dd < NERs) ? ts_ner[b * NERs + dd] : 0.f;
      Xh[b * KPAD + 2 * HIDs + dd] = hv;
      Xt[b * KPAD + 2 * HIDs + dd] = tv;
    }
  }
}

// ---------------------------------------------------------------------------
// Kernel 4: Y = tanh(X @ W^T + bias) via V_WMMA_F32_16X16X4_F32
//   X: (16, KPAD) padded, W: (768, KCAT), Y: (16, 768)
//   8 waves per 16x16 tile; uniform 48 chunks/wave, depth-PF prefetch, LDS-reduce.
// ---------------------------------------------------------------------------
__global__ void k_gemm_tanh(const float* __restrict__ Xh, const float* __restrict__ Xt,
                            const float* __restrict__ Wh, const float* __restrict__ bh,
                            const float* __restrict__ Wt, const float* __restrict__ bt,
                            float* __restrict__ hs2, float* __restrict__ ts2) {
  int tile = blockIdx.x;           // 0..95
  int sel  = tile / 48;
  int nt   = tile % 48;
  const float* X    = sel ? Xt : Xh;
  const float* W    = sel ? Wt : Wh;
  const float* bias = sel ? bt : bh;
  float* Y          = sel ? ts2 : hs2;

  int lane = threadIdx.x & 31;
  int w    = threadIdx.x >> 5;     // 0..7 wave id
  int half = lane >> 4;            // 0: K=0,1  1: K=2,3 of each chunk
  int lr   = lane & 15;
  const float* arow = X + lr * KPAD + 2 * half + 4 * w;             // chunk c: +32*i
  const float* brow = W + (nt * 16 + lr) * KCAT + 2 * half + 4 * w;

  const int CPW = 384 / KWAVES;    // 48 uniform chunks per wave (k0 < 1536)
  v8f acc = {};
  v2f abuf[PF], bbuf[PF];
#pragma unroll
  for (int p = 0; p < PF; ++p) {
    abuf[p] = *(const v2f*)(arow + 32 * p);
    bbuf[p] = *(const v2f*)(brow + 32 * p);
  }
  for (int i = PF; i < CPW; i += PF) {     // 11 refill stages, 4 pairs in flight
#pragma unroll
    for (int p = 0; p < PF; ++p) {
      acc = __builtin_amdgcn_wmma_f32_16x16x4_f32(false, abuf[p], false, bbuf[p],
                                                  (short)0, acc, false, false);
      abuf[p] = *(const v2f*)(arow + 32 * (i + p));
      bbuf[p] = *(const v2f*)(brow + 32 * (i + p));
    }
  }
#pragma unroll
  for (int p = 0; p < PF; ++p)
    acc = __builtin_amdgcn_wmma_f32_16x16x4_f32(false, abuf[p], false, bbuf[p],
                                                (short)0, acc, false, false);

  if (w == 0) {  // wave 0 mops up chunks 384 (k0=1536, valid) and 385 (guarded)
    const float* a0 = X + lr * KPAD + 2 * half;
    const float* b0 = W + (nt * 16 + lr) * KCAT + 2 * half;
    v2f a  = *(const v2f*)(a0 + 1536);
    v2f b2 = *(const v2f*)(b0 + 1536);
    acc = __builtin_amdgcn_wmma_f32_16x16x4_f32(false, a, false, b2,
                                                (short)0, acc, false, false);
    v2f bz = {};
    v2f a3 = *(const v2f*)(a0 + 1540);              // X pad cols 1542/1543 are 0
    v2f b3 = (half == 0) ? *(const v2f*)(b0 + 1540) : bz;  // W row ends at 1542
    acc = __builtin_amdgcn_wmma_f32_16x16x4_f32(false, a3, false, b3,
                                                (short)0, acc, false, false);
  }

  __shared__ float red[KWAVES * 32 * 8];
#pragma unroll
  for (int r = 0; r < 8; ++r) red[(w * 32 + lane) * 8 + r] = acc[r];
  __syncthreads();
  if (w == 0) {
    int col = nt * 16 + lr;
    float bv = bias[col];
#pragma unroll
    for (int r = 0; r < 8; ++r) {
      float s = 0.f;
#pragma unroll
      for (int q = 0; q < KWAVES; ++q) s += red[(q * 32 + lane) * 8 + r];
      int row = r + (half ? 8 : 0);  // D layout: lanes 0-15 -> M=r, 16-31 -> M=r+8
      Y[row * HIDs + col] = tanhf(s + bv);
    }
  }
}

// ---------------------------------------------------------------------------
// Kernel 5: block bilinear features bl[b, i*64 + j*8 + k] = hs2[b,i*8+j]*ts2[b,i*8+k]
// ---------------------------------------------------------------------------
__global__ void k_bl(const float* __restrict__ hs2, const float* __restrict__ ts2,
                     float* __restrict__ bl) {
  int idx = blockIdx.x * blockDim.x + threadIdx.x;  // 16*6144
  if (idx >= Bsz * KBL) return;
  int t = idx % KBL;
  int b = idx / KBL;
  int i = t / (BLKs * BLKs);
  int j = (t / BLKs) % BLKs;
  int k = t % BLKs;
  bl[idx] = hs2[b * HIDs + i * BLKs + j] * ts2[b * HIDs + i * BLKs + k];
}

// ---------------------------------------------------------------------------
// Kernel 6: logits = bl @ Wb^T + bb via WMMA. N=97 -> 7 tiles, invalid cols = 0.
//   8 waves per tile; uniform 192 chunks/wave, depth-PF prefetch, LDS-reduce.
// ---------------------------------------------------------------------------
__global__ void k_logits(const float* __restrict__ bl, const float* __restrict__ Wb,
                         const float* __restrict__ bb, float* __restrict__ out) {
  int nt   = blockIdx.x;           // 0..6
  int lane = threadIdx.x & 31;
  int w    = threadIdx.x >> 5;     // 0..7
  int half = lane >> 4;
  int lr   = lane & 15;
  int n    = nt * 16 + lr;
  bool nvalid = (n < NCLSs);
  const float* arow = bl + lr * KBL + 2 * half + 4 * w;
  const float* brow = Wb + (long)(nvalid ? n : 0) * KBL + 2 * half + 4 * w;

  const int CPW = (KBL / 4) / KWAVES;   // 192 chunks per wave
  v8f acc = {};
  v2f bz = {};
  v2f abuf[PF], bbuf[PF];
#pragma unroll
  for (int p = 0; p < PF; ++p) {
    abuf[p] = *(const v2f*)(arow + 32 * p);
    bbuf[p] = nvalid ? *(const v2f*)(brow + 32 * p) : bz;
  }
  for (int i = PF; i < CPW; i += PF) {     // 47 refill stages
#pragma unroll
    for (int p = 0; p < PF; ++p) {
      acc = __builtin_amdgcn_wmma_f32_16x16x4_f32(false, abuf[p], false, bbuf[p],
                                                  (short)0, acc, false, false);
      abuf[p] = *(const v2f*)(arow + 32 * (i + p));
      bbuf[p] = nvalid ? *(const v2f*)(brow + 32 * (i + p)) : bz;
    }
  }
#pragma unroll
  for (int p = 0; p < PF; ++p)
    acc = __builtin_amdgcn_wmma_f32_16x16x4_f32(false, abuf[p], false, bbuf[p],
                                                (short)0, acc, false, false);

  __shared__ float red[KWAVES * 32 * 8];
#pragma unroll
  for (int r = 0; r < 8; ++r) red[(w * 32 + lane) * 8 + r] = acc[r];
  __syncthreads();
  if (w == 0 && nvalid) {
    float bv = bb[n];
#pragma unroll
    for (int r = 0; r < 8; ++r) {
      float s = 0.f;
#pragma unroll
      for (int q = 0; q < KWAVES; ++q) s += red[(q * 32 + lane) * 8 + r];
      int row = r + (half ? 8 : 0);
      out[row * NCLSs + n] = s + bv;
    }
  }
}

// ---------------------------------------------------------------------------
extern "C" void kernel_launch(void* const* d_in, const int* in_sizes, int n_in,
                              void* d_out, int out_size, void* d_ws, size_t ws_size,
                              hipStream_t stream) {
  const float* seq     = (const float*)d_in[0];   // (16,512,768)
  const float* attn    = (const float*)d_in[1];   // (16,12,512,512)
  const int*   epos    = (const int*)  d_in[2];   // (16,2,4)
  const float* hs_ner  = (const float*)d_in[3];   // (16,6)
  const float* ts_ner  = (const float*)d_in[4];   // (16,6)
  const float* Wh      = (const float*)d_in[5];   // (768,1542)
  const float* bh      = (const float*)d_in[6];   // (768,)
  const float* Wt      = (const float*)d_in[7];   // (768,1542)
  const float* bt      = (const float*)d_in[8];   // (768,)
  const float* Wb      = (const float*)d_in[9];   // (97,6144)
  const float* bb      = (const float*)d_in[10];  // (97,)
  float* out = (float*)d_out;                     // (16,97)

  float* ws  = (float*)d_ws;
  float* Xh  = ws;                    // 16*1544
  float* Xt  = Xh  + Bsz * KPAD;      // 16*1544
  float* htn = Xt  + Bsz * KPAD;      // 16*512
  float* hs2 = htn + Bsz * Lsz;       // 16*768
  float* ts2 = hs2 + Bsz * HIDs;      // 16*768
  float* blw = ts2 + Bsz * HIDs;      // 16*6144

  k_entity_emb<<<(Bsz * 2 * HIDs + 255) / 256, 256, 0, stream>>>(seq, epos, Xh, Xt);
  k_ht_att<<<Bsz, 256, 0, stream>>>(attn, epos, htn);
  k_rs<<<(Bsz * (HIDs / 2) + 255) / 256, 256, 0, stream>>>(seq, htn, hs_ner, ts_ner, Xh, Xt);
  k_gemm_tanh<<<96, 256, 0, stream>>>(Xh, Xt, Wh, bh, Wt, bt, hs2, ts2);
  k_bl<<<(Bsz * KBL + 255) / 256, 256, 0, stream>>>(hs2, ts2, blw);
  k_logits<<<7, 256, 0, stream>>>(blw, Wb, bb, out);
}